// SagePolyConv_23845658427616
// MI455X (gfx1250) — compile-verified
//
#include <hip/hip_runtime.h>
#include <cstdint>
#include <cstddef>

// SagePolyConv on MI455X (gfx1250).
// h = sum_k theta_k * feat_k,  feat_k = feat_{k-1} - Dinv^{1/2} A Dinv^{1/2} feat_{k-1}
// Memory/atomic bound (~0.02 FLOP/byte); feat (25.6MB) is L2-resident (192MB L2).
// CDNA5 path: per-edge 256B feature-row gathers via GLOBAL_LOAD_ASYNC_TO_LDS_B64
// (32 lanes x 8B = one row), 4-deep ring per wave, s_wait_asynccnt-synchronized.
// 16K waves x 4 in-flight gathers ≈ 65K outstanding requests -> bandwidth-bound.

#define D_FEAT 64
#define SC_WAVES 8   // waves per block in scatter kernel (256 threads)
#define PIPE 4       // async pipeline depth per wave

static constexpr float THETA0 = 1.0f;

// ---------------- setup kernels ----------------

__global__ void zero_f32(float* __restrict__ p, int n) {
  int i = blockIdx.x * blockDim.x + threadIdx.x;
  int stride = gridDim.x * blockDim.x;
  for (; i < n; i += stride) p[i] = 0.0f;
}

__global__ void degree_kernel(const int* __restrict__ dst,
                              float* __restrict__ deg, int n_edges) {
  int i = blockIdx.x * blockDim.x + threadIdx.x;
  int stride = gridDim.x * blockDim.x;
  for (; i < n_edges; i += stride) {
    __builtin_prefetch(&dst[i + 4 * stride], 0, 0);   // global_prefetch_b8
    atomicAdd(&deg[dst[i]], 1.0f);
  }
}

// in place: deg -> clip(deg,1)^(-1/2)
__global__ void dinv_kernel(float* __restrict__ deg, int n) {
  int i = blockIdx.x * blockDim.x + threadIdx.x;
  int stride = gridDim.x * blockDim.x;
  for (; i < n; i += stride) {
    float d = deg[i];
    d = d < 1.0f ? 1.0f : d;
    deg[i] = 1.0f / sqrtf(d);
  }
}

// fc = feat ; h = theta0 * feat   (vectorized float4 over N*D)
__global__ void init_kernel(const float* __restrict__ feat,
                            float* __restrict__ fc,
                            float* __restrict__ h, int n4) {
  int i = blockIdx.x * blockDim.x + threadIdx.x;
  int stride = gridDim.x * blockDim.x;
  const float4* f4 = (const float4*)feat;
  float4* fc4 = (float4*)fc;
  float4* h4 = (float4*)h;
  for (; i < n4; i += stride) {
    float4 v = f4[i];
    fc4[i] = v;
    h4[i] = make_float4(THETA0 * v.x, THETA0 * v.y, THETA0 * v.z, THETA0 * v.w);
  }
}

// ---------------- scatter: agg[dst] += fc[src] * dinv[src] ----------------
// One wave per edge step: 32 lanes x float2 = one 64-float row.
// 4-deep ring of async row-gathers into LDS; dinv[src] fetched at issue time
// and rotated through the pipeline so its random-load latency is hidden too.

#define ISSUE_ASYNC(ldsaddr, node)                                         \
  do {                                                                     \
    uint32_t goff_ = (uint32_t)(node) * (D_FEAT * 4u) + lane_off;          \
    asm volatile("global_load_async_to_lds_b64 %0, %1, %2"                 \
                 :: "v"(ldsaddr), "v"(goff_), "s"(fc) : "memory");         \
  } while (0)

__global__ void scatter_kernel(const float* __restrict__ fc,
                               const float* __restrict__ dinv,
                               const int* __restrict__ src,
                               const int* __restrict__ dst,
                               float* __restrict__ agg,
                               int n_edges) {
  __shared__ __align__(16) float s_rows[SC_WAVES * PIPE * D_FEAT];

  const int lane  = threadIdx.x & 31;
  const int wave  = threadIdx.x >> 5;
  const int gwave = blockIdx.x * SC_WAVES + wave;
  const int nwaves = gridDim.x * SC_WAVES;
  const uint32_t lane_off = (uint32_t)lane * 8u;

  // this lane's 8-byte chunk in each of the 4 ring slots
  float* base = &s_rows[wave * PIPE * D_FEAT];
  const float* r0 = base + 0 * D_FEAT + lane * 2;
  const float* r1 = base + 1 * D_FEAT + lane * 2;
  const float* r2 = base + 2 * D_FEAT + lane * 2;
  const float* r3 = base + 3 * D_FEAT + lane * 2;
  // per-lane LDS byte addresses (low 32 bits of flat addr == LDS offset)
  uint32_t a0 = (uint32_t)(uintptr_t)r0;
  uint32_t a1 = (uint32_t)(uintptr_t)r1;
  uint32_t a2 = (uint32_t)(uintptr_t)r2;
  uint32_t a3 = (uint32_t)(uintptr_t)r3;

  int e = gwave;
  if (e >= n_edges) return;            // wave-uniform: EXEC stays full below

  // prime up to 3 stages (all branches wave-uniform)
  float w0 = 0.0f, w1 = 0.0f, w2 = 0.0f, w3 = 0.0f;
  {
    int sn = src[e];
    w0 = dinv[sn];
    ISSUE_ASYNC(a0, sn);
  }
  if (e + nwaves < n_edges) {
    int sn = src[e + nwaves];
    w1 = dinv[sn];
    ISSUE_ASYNC(a1, sn);
  }
  if (e + 2 * nwaves < n_edges) {
    int sn = src[e + 2 * nwaves];
    w2 = dinv[sn];
    ISSUE_ASYNC(a2, sn);
  }

  for (; e < n_edges; e += nwaves) {
    int e1 = e + nwaves, e2 = e + 2 * nwaves, e3 = e + 3 * nwaves;
    if (e3 < n_edges) {
      __builtin_prefetch(&src[e + 6 * nwaves], 0, 0);   // stream edge indices
      __builtin_prefetch(&dst[e + 6 * nwaves], 0, 0);
      int sn = src[e3];
      w3 = dinv[sn];
      ISSUE_ASYNC(a3, sn);
      // 4 in flight; async loads retire in order -> oldest row has landed
      asm volatile("s_wait_asynccnt 3" ::: "memory");
    } else if (e2 < n_edges) {
      asm volatile("s_wait_asynccnt 2" ::: "memory");
    } else if (e1 < n_edges) {
      asm volatile("s_wait_asynccnt 1" ::: "memory");
    } else {
      asm volatile("s_wait_asynccnt 0" ::: "memory");
    }

    float2 v = *(const float2*)r0;
    // ensure v is in VGPRs before this slot is recycled by the next issue
    asm volatile("s_wait_dscnt 0" ::: "memory");

    int dvtx = dst[e];
    float* dp = agg + (size_t)dvtx * D_FEAT + lane * 2;
    atomicAdd(dp,     v.x * w0);      // global_atomic_add_f32, L2-resident
    atomicAdd(dp + 1, v.y * w0);

    // rotate the ring
    { uint32_t t = a0; a0 = a1; a1 = a2; a2 = a3; a3 = t; }
    { const float* t = r0; r0 = r1; r1 = r2; r2 = r3; r3 = t; }
    w0 = w1; w1 = w2; w2 = w3;
  }
}

// ---------------- fused update ----------------
// fc = fc - agg*dinv[row] ; h += theta*fc ; agg = 0 (ready for next hop)

__global__ void update_kernel(float* __restrict__ fc,
                              float* __restrict__ agg,
                              const float* __restrict__ dinv,
                              float* __restrict__ h,
                              float theta, int n_rows) {
  int i = blockIdx.x * blockDim.x + threadIdx.x;
  int stride = gridDim.x * blockDim.x;
  int total = n_rows * (D_FEAT / 4);        // float4 elements
  float4* fc4 = (float4*)fc;
  float4* ag4 = (float4*)agg;
  float4* h4  = (float4*)h;
  for (; i < total; i += stride) {
    int row = i >> 4;                       // 16 float4 per 64-wide row
    float w = dinv[row];
    float4 a = ag4[i];
    float4 f = fc4[i];
    f.x -= a.x * w; f.y -= a.y * w; f.z -= a.z * w; f.w -= a.w * w;
    fc4[i] = f;
    float4 hh = h4[i];
    hh.x += theta * f.x; hh.y += theta * f.y;
    hh.z += theta * f.z; hh.w += theta * f.w;
    h4[i] = hh;
    ag4[i] = make_float4(0.0f, 0.0f, 0.0f, 0.0f);   // re-zero for next hop
  }
}

// ---------------- launcher ----------------

extern "C" void kernel_launch(void* const* d_in, const int* in_sizes, int n_in,
                              void* d_out, int out_size, void* d_ws, size_t ws_size,
                              hipStream_t stream) {
  const float* feat = (const float*)d_in[0];   // [N, 64]
  const int*   src  = (const int*)d_in[1];     // [E]
  const int*   dst  = (const int*)d_in[2];     // [E]
  float*       h    = (float*)d_out;           // [N, 64]

  const int N = in_sizes[0] / D_FEAT;
  const int E = in_sizes[1];

  // workspace layout: deg/dinv [N] | agg [N*64] | fc [N*64]  (~52 MB)
  float* deg = (float*)d_ws;
  float* agg = deg + N;
  float* fcur = agg + (size_t)N * D_FEAT;

  const float theta[4] = {1.0f, 0.8f, 0.5f, 0.3f};

  // zero deg + agg (contiguous)
  zero_f32<<<2048, 256, 0, stream>>>(deg, N * (D_FEAT + 1));
  degree_kernel<<<1024, 256, 0, stream>>>(dst, deg, E);
  dinv_kernel<<<(N + 255) / 256, 256, 0, stream>>>(deg, N);
  init_kernel<<<2048, 256, 0, stream>>>(feat, fcur, h, N * (D_FEAT / 4));

  for (int k = 1; k < 4; ++k) {
    scatter_kernel<<<2048, 256, 0, stream>>>(fcur, deg, src, dst, agg, E);
    update_kernel<<<2048, 256, 0, stream>>>(fcur, agg, deg, h, theta[k], N);
  }
}